// OnlineSmoothingCELoss_47021301957460
// MI455X (gfx1250) — compile-verified
//
#include <hip/hip_runtime.h>

typedef __attribute__((ext_vector_type(2))) float v2f;
typedef __attribute__((ext_vector_type(8))) float v8f;

#define NCLS 15
#define TPB  256
#define WPB  8   // waves per block (wave32)

__global__ __launch_bounds__(TPB) void osce_init(float* __restrict__ out,
                                                 float* __restrict__ ws) {
  int t = blockIdx.x * blockDim.x + threadIdx.x;
  if (t < 241) out[t] = 0.0f;   // loss + 15x15 stats + 15 counter
  if (t < 2)   ws[t]  = 0.0f;   // loss_sum, valid_count
}

__global__ __launch_bounds__(TPB) void osce_main(
    const float* __restrict__ y_pred, const int* __restrict__ y_true,
    const float* __restrict__ cw, const float* __restrict__ sm,
    float* __restrict__ out, float* __restrict__ ws, int n) {
  __shared__ float xbuf[WPB][480];      // coalesced staging of 32 rows/wave
  __shared__ float btile[WPB][32][17];  // B rows: probs[0..14], 1.0 (pad 17)
  __shared__ float acorr[WPB][32];
  __shared__ int   aidx[WPB][32];
  __shared__ float wsm[NCLS * 16];      // class_weight[c]*soft_matrix[k][c]
  __shared__ float sstat[16][16];       // block-level WMMA acc reduction
  __shared__ float red[TPB];

  const int tid  = threadIdx.x;
  const int lane = tid & 31;
  const int wv   = tid >> 5;
  const int half = lane >> 4;
  const int l15  = lane & 15;

  for (int i = tid; i < NCLS * NCLS; i += TPB)
    wsm[(i / NCLS) * 16 + (i % NCLS)] = cw[i % NCLS] * sm[i];
  sstat[tid >> 4][tid & 15] = 0.0f;
  __syncthreads();

  const int nWaves = (int)gridDim.x * WPB;
  const int gw     = (int)blockIdx.x * WPB + wv;
  const size_t total = (size_t)n * NCLS;

  v8f acc = {};
  float lsum = 0.0f, vcnt = 0.0f;

  for (long base = (long)gw * 32; base < (long)n; base += (long)nWaves * 32) {
    // ---- fully coalesced load of 32 rows (480 dwords) into LDS ----
    const size_t rb = (size_t)base * NCLS;
#pragma unroll
    for (int j = 0; j < NCLS; ++j) {
      size_t off = rb + (size_t)j * 32 + (size_t)lane;
      xbuf[wv][j * 32 + lane] = (off < total) ? y_pred[off] : 0.0f;
    }
    asm volatile("s_wait_dscnt 0" ::: "memory");

    float x[NCLS];
#pragma unroll
    for (int c = 0; c < NCLS; ++c) x[c] = xbuf[wv][lane * NCLS + c];

    const long r   = base + lane;
    const bool inb = (r < (long)n);
    const int  yt  = inb ? y_true[r] : -1;
    const int  yidx = (yt < 0) ? 0 : yt;

    // log-softmax + argmax (first-max tie-break, ascending strict >)
    float mx = x[0]; int am = 0;
#pragma unroll
    for (int c = 1; c < NCLS; ++c) if (x[c] > mx) { mx = x[c]; am = c; }
    float p[NCLS]; float se = 0.0f;
#pragma unroll
    for (int c = 0; c < NCLS; ++c) { p[c] = expf(x[c] - mx); se += p[c]; }
    const float inv  = 1.0f / se;
    const float lse  = logf(se);
    const float corr = (am == yt) ? 1.0f : 0.0f;  // yt==-1 never matches

    // ---- stage WMMA operands: B = [probs | 1.0], A = onehot(yidx)*corr ----
#pragma unroll
    for (int c = 0; c < NCLS; ++c) btile[wv][lane][c] = p[c] * inv;
    btile[wv][lane][NCLS] = 1.0f;   // col 15 accumulates counter_delta
    acorr[wv][lane] = corr;
    aidx[wv][lane]  = yidx;
    asm volatile("s_wait_dscnt 0" ::: "memory");

    // stats[k,c] += sum_rows correct * probs  via onehot^T x probs
    // A 16x4 f32 layout: lanes0-15 hold K0/K1, lanes16-31 K2/K3 (M=lane&15)
    // B 4x16  f32 layout: VGPR0 = K0|K2 across lane halves, VGPR1 = K1|K3
#pragma unroll
    for (int q = 0; q < 8; ++q) {
      const int k0 = q * 4 + half * 2;
      v2f a, b;
      b.x = btile[wv][k0][l15];
      b.y = btile[wv][k0 + 1][l15];
      const int   i0 = aidx[wv][k0],      i1 = aidx[wv][k0 + 1];
      const float c0 = acorr[wv][k0],     c1 = acorr[wv][k0 + 1];
      a.x = (i0 == l15) ? c0 : 0.0f;
      a.y = (i1 == l15) ? c1 : 0.0f;
      acc = __builtin_amdgcn_wmma_f32_16x16x4_f32(false, a, false, b,
                                                  (short)0, acc, false, false);
    }
    asm volatile("s_wait_dscnt 0" ::: "memory");  // WAR vs next iter writes

    // weighted smoothed-CE: -sum_c w[c]*soft[yidx][c]*logp[c], valid rows only
    if (yt >= 0) {
      float s = 0.0f;
#pragma unroll
      for (int c = 0; c < NCLS; ++c) s += wsm[yidx * 16 + c] * (x[c] - mx - lse);
      lsum -= s;
      vcnt += 1.0f;
    }
  }

  // ---- block reduction of WMMA accumulators (C/D layout: VGPR j -> M=j|j+8) ----
  __syncthreads();
  {
    const int mb = half * 8;
#pragma unroll
    for (int j = 0; j < 8; ++j) atomicAdd(&sstat[j + mb][l15], acc[j]);
  }
  __syncthreads();
  {
    const int M = tid >> 4, Nn = tid & 15;
    const float v = sstat[M][Nn];
    if (M < NCLS && v != 0.0f) {
      if (Nn < NCLS) atomicAdd(&out[1 + M * NCLS + Nn], v);   // stats_delta
      else           atomicAdd(&out[1 + NCLS * NCLS + M], v); // counter_delta
    }
  }

  // ---- loss / valid-count block reduction ----
  red[tid] = lsum;
  __syncthreads();
  for (int s = TPB / 2; s > 0; s >>= 1) {
    if (tid < s) red[tid] += red[tid + s];
    __syncthreads();
  }
  if (tid == 0) atomicAdd(&ws[0], red[0]);
  __syncthreads();
  red[tid] = vcnt;
  __syncthreads();
  for (int s = TPB / 2; s > 0; s >>= 1) {
    if (tid < s) red[tid] += red[tid + s];
    __syncthreads();
  }
  if (tid == 0) atomicAdd(&ws[1], red[0]);
}

__global__ void osce_fin(float* __restrict__ out, const float* __restrict__ ws) {
  out[0] = ws[0] / ws[1];
}

extern "C" void kernel_launch(void* const* d_in, const int* in_sizes, int n_in,
                              void* d_out, int out_size, void* d_ws, size_t ws_size,
                              hipStream_t stream) {
  const float* y_pred = (const float*)d_in[0];
  const int*   y_true = (const int*)d_in[1];
  const float* cw     = (const float*)d_in[2];
  const float* sm     = (const float*)d_in[3];
  float* out = (float*)d_out;
  float* ws  = (float*)d_ws;
  const int n = in_sizes[1];   // number of rows (y_true count)

  osce_init<<<1, TPB, 0, stream>>>(out, ws);
  const int blocks = 1024;     // 8192 waves: enough to saturate HBM, cheap tail
  osce_main<<<blocks, TPB, 0, stream>>>(y_pred, y_true, cw, sm, out, ws, n);
  osce_fin<<<1, 1, 0, stream>>>(out, ws);
}